// LocationAttention_58660663329112
// MI455X (gfx1250) — compile-verified
//
#include <hip/hip_runtime.h>

// LocationAttention scan on MI455X (gfx1250):
//  Exclusive prefix sums computed as triangular-matrix WMMAs
//  (V_WMMA_F32_16X16X4_F32, full fp32 precision). Carry is kept OUT of the
//  WMMA accumulator: C starts at 0 and the running carry is added at store
//  time, so consecutive 16-row blocks' WMMA chains are independent and the
//  only serial dependency is one scalar add after a ds_bpermute.
//  Two-pass chunked scan; x is 16 MB so the second pass re-reads it from the
//  192 MB L2, keeping HBM traffic ~= in + out (~32 MB @ 23.3 TB/s ~ 1.4 us).

#define BB 8
#define SS 2048
#define HH 256
#define CHUNK 128
#define NCH (SS / CHUNK) // 16 chunks per batch -> 128 workgroups per stage

typedef __attribute__((ext_vector_type(2))) float v2f;
typedef __attribute__((ext_vector_type(8))) float v8f;

// ---------------------------------------------------------------------------
// Stage 1: per (batch, chunk): a[s] = exp(x[s].W + b); chunk sums of a and a*x
// ---------------------------------------------------------------------------
__global__ __launch_bounds__(256) void la_stage1(
    const float* __restrict__ X, const float* __restrict__ W,
    const float* __restrict__ bias, float* __restrict__ aArr,
    float* __restrict__ aSum, float* __restrict__ xSum) {
  __shared__ float part[8 * HH]; // per-wave partial channel sums
  __shared__ float aPart[8];

  const int bidx = blockIdx.x / NCH;
  const int cidx = blockIdx.x % NCH;
  const int chunkBase = cidx * CHUNK;
  const int lane = threadIdx.x & 31;
  const int wv = threadIdx.x >> 5; // 8 waves, CHUNK/8 = 16 positions each

  // W cached per lane: channels 4*lane..+3 and 128+4*lane..+3
  const float4 wa = ((const float4*)W)[lane];
  const float4 wb = ((const float4*)W)[32 + lane];
  const float bscal = bias[0];

  float acc[8] = {0.f, 0.f, 0.f, 0.f, 0.f, 0.f, 0.f, 0.f};
  float accA = 0.f;

  for (int i = 0; i < CHUNK / 8; ++i) {
    const int pos = chunkBase + wv * (CHUNK / 8) + i;
    const float* xrow = X + ((size_t)bidx * SS + pos) * HH;
    const float4 xa = ((const float4*)xrow)[lane];
    const float4 xb = ((const float4*)xrow)[32 + lane];
    float p = xa.x * wa.x + xa.y * wa.y + xa.z * wa.z + xa.w * wa.w +
              xb.x * wb.x + xb.y * wb.y + xb.z * wb.z + xb.w * wb.w;
    // wave32 xor-reduction: every lane ends with the full dot product
    for (int off = 16; off > 0; off >>= 1) p += __shfl_xor(p, off, 32);
    const float a = expf(p + bscal);
    if (lane == 0) aArr[(size_t)bidx * SS + pos] = a;
    accA += a;
    acc[0] += a * xa.x; acc[1] += a * xa.y; acc[2] += a * xa.z; acc[3] += a * xa.w;
    acc[4] += a * xb.x; acc[5] += a * xb.y; acc[6] += a * xb.z; acc[7] += a * xb.w;
  }

  for (int jj = 0; jj < 4; ++jj) {
    part[wv * HH + 4 * lane + jj] = acc[jj];
    part[wv * HH + 128 + 4 * lane + jj] = acc[4 + jj];
  }
  if (lane == 0) aPart[wv] = accA;
  __syncthreads();

  if (threadIdx.x < HH) { // deterministic cross-wave combine (fixed order)
    float s = 0.f;
    for (int w = 0; w < 8; ++w) s += part[w * HH + threadIdx.x];
    xSum[((size_t)bidx * NCH + cidx) * HH + threadIdx.x] = s;
  }
  if (threadIdx.x == 0) {
    float s = 0.f;
    for (int w = 0; w < 8; ++w) s += aPart[w];
    aSum[bidx * NCH + cidx] = s;
  }
}

// ---------------------------------------------------------------------------
// Stage 2: per (batch, chunk): exclusive scans via L(16x16, strict lower ones)
// matmuls on V_WMMA_F32_16X16X4_F32. 16 waves x 16 channels; each wave walks
// CHUNK/16 blocks of 16 positions. C accumulator starts at 0; the running
// carry is applied at store time to keep WMMA chains independent per block.
// ---------------------------------------------------------------------------
__global__ __launch_bounds__(512) void la_stage2(
    const float* __restrict__ X, const float* __restrict__ aArr,
    const float* __restrict__ aSum, const float* __restrict__ xSum,
    float* __restrict__ OUT) {
  __shared__ float sA[CHUNK]; // a values for this chunk
  __shared__ float sR[CHUNK]; // 1 / (aCarry + excl_a + eps)  per position

  const int bidx = blockIdx.x / NCH;
  const int cidx = blockIdx.x % NCH;
  const int chunkBase = cidx * CHUNK;
  const int lane = threadIdx.x & 31;
  const int wv = threadIdx.x >> 5; // 0..15 -> channel tile
  const int n = lane & 15;         // column index (channel within tile)
  const int half = lane >> 4;      // lane half selects row groups

  // Wave 0: exclusive scan of a[chunk] + cross-chunk aCarry, folded into a
  // per-position reciprocal. Exact IEEE division happens only here (cold).
  if (threadIdx.x < 32) {
    float aCarry = 0.f;
    for (int cc = 0; cc < cidx; ++cc) aCarry += aSum[bidx * NCH + cc];
    float run = 0.f;
    for (int i = 0; i < CHUNK / 32; ++i) {
      float v = aArr[(size_t)bidx * SS + chunkBase + i * 32 + lane];
      sA[i * 32 + lane] = v;
      float incl = v;
      for (int off = 1; off < 32; off <<= 1) {
        float nv = __shfl_up(incl, off, 32);
        if (lane >= off) incl += nv;
      }
      float excl = __shfl_up(incl, 1, 32);
      if (lane == 0) excl = 0.f;
      sR[i * 32 + lane] = 1.0f / (aCarry + run + excl + 1e-10f);
      run += __shfl(incl, 31, 32);
    }
  }
  __syncthreads();

  const int ch = wv * 16 + n;
  // Cross-chunk numerator carry (exclusive over previous chunks).
  float numCarry = 0.f;
  for (int cc = 0; cc < cidx; ++cc)
    numCarry += xSum[((size_t)bidx * NCH + cc) * HH + ch];

  // A operand: strict lower-triangular ones, split into four 16x4 K-slices.
  // f32 A 16x4 layout: m = lane&15; vgpr j holds K = j + 2*(lane>=16).
  // Pin the 8 values in VGPRs (opaque producer) so the unrolled loop does not
  // rematerialize v_cmp/v_cndmask per block.
  float Aa[4][2];
  for (int k = 0; k < 4; ++k)
    for (int j = 0; j < 2; ++j) {
      float v = ((4 * k + j + 2 * half) < n) ? 1.f : 0.f;
      asm("" : "+v"(v)); // keep loop-invariant, non-rematerializable
      Aa[k][j] = v;
    }

  const float* xbase = X + ((size_t)bidx * SS + chunkBase) * HH + ch;
  float* obase = OUT + ((size_t)bidx * SS + chunkBase) * HH + ch;

  for (int t = 0; t < CHUNK / 16; ++t) {
    // B operand: M[row][n] = a[row]*x[row][ch]; B 4x16 layout:
    // vgpr j holds row (4k + j + 2*half), col = n.
    float Bv[4][2];
    for (int k = 0; k < 4; ++k)
      for (int j = 0; j < 2; ++j) {
        const int lr = t * 16 + 4 * k + j + 2 * half;
        Bv[k][j] = sA[lr] * xbase[(size_t)lr * HH];
      }

    // Carry-free block prefix: D = L x M  (C starts at zero).
    v8f cacc = {0.f, 0.f, 0.f, 0.f, 0.f, 0.f, 0.f, 0.f};
    for (int k = 0; k < 4; ++k) {
      v2f Av; Av[0] = Aa[k][0]; Av[1] = Aa[k][1];
      v2f Bw; Bw[0] = Bv[k][0]; Bw[1] = Bv[k][1];
      cacc = __builtin_amdgcn_wmma_f32_16x16x4_f32(
          /*neg_a=*/false, Av, /*neg_b=*/false, Bw,
          /*c_mod=*/(short)0, cacc, /*reuse_a=*/false, /*reuse_b=*/false);
    }

    // out[p] = (carry + D[p][n]) * sR[p]  -- carry applied outside the WMMA
    for (int jj = 0; jj < 8; ++jj) {
      const int lr = t * 16 + jj + 8 * half; // C/D vgpr jj holds rows jj, jj+8
      obase[(size_t)lr * HH] = (numCarry + cacc[jj]) * sR[lr];
    }

    // Column sum of this block's M = D[15][n] + M[15][n] (lanes 16..31:
    // C vgpr7 and B3 vgpr1); broadcast and fold into the scalar carry chain.
    const float colsum = __shfl(cacc[7] + Bv[3][1], 16 + n, 32);
    numCarry += colsum;
  }
}

// ---------------------------------------------------------------------------
extern "C" void kernel_launch(void* const* d_in, const int* in_sizes, int n_in,
                              void* d_out, int out_size, void* d_ws, size_t ws_size,
                              hipStream_t stream) {
  (void)in_sizes; (void)n_in; (void)out_size; (void)ws_size;
  const float* X = (const float*)d_in[0];    // [8,2048,256] f32
  const float* W = (const float*)d_in[1];    // [256,1] f32
  const float* bias = (const float*)d_in[2]; // [1] f32
  float* OUT = (float*)d_out;                // [8,2048,256] f32

  float* ws = (float*)d_ws;
  float* aArr = ws;                 // BB*SS      = 16384 floats
  float* aSum = ws + BB * SS;       // BB*NCH     = 128 floats
  float* xSum = aSum + BB * NCH;    // BB*NCH*HH  = 32768 floats (~193 KB total)

  la_stage1<<<dim3(BB * NCH), dim3(256), 0, stream>>>(X, W, bias, aArr, aSum, xSum);
  la_stage2<<<dim3(BB * NCH), dim3(512), 0, stream>>>(X, aArr, aSum, xSum, OUT);
}